// PointToSupernodeMessagePassing_65781719106288
// MI455X (gfx1250) — compile-verified
//
#include <hip/hip_runtime.h>
#include <math.h>

typedef float v2f __attribute__((ext_vector_type(2)));
typedef float v8f __attribute__((ext_vector_type(8)));

#define BB    8
#define NN    16384
#define MM    1024
#define KN    32
#define DD    256
#define DIN   260        // d + 4 (rel xyz + dist)
#define SSN   4          // supernodes per workgroup
#define RR    (SSN*KN)   // 128 edge rows per workgroup
#define LSTR  264        // padded LDS row stride (floats)
#define NWAVE 16
#define LN_EPS 1e-5f

__global__ __launch_bounds__(512) void p2s_msg_kernel(
    const float* __restrict__ point_feat,     // [B,N,256]
    const float* __restrict__ point_xyz,      // [B,N,3]
    const float* __restrict__ supernode_xyz,  // [B,M,3]
    const int*   __restrict__ neighbor_idx,   // [B,M,32]
    const unsigned char* __restrict__ neighbor_mask, // [B,M,32] (bool)
    const float* __restrict__ init_feat,      // [B,M,256]
    const float* __restrict__ W1,             // [260,256]
    const float* __restrict__ b1,             // [256]
    const float* __restrict__ W2,             // [256,256]
    const float* __restrict__ b2,             // [256]
    const float* __restrict__ ln_g,           // [256]
    const float* __restrict__ ln_b,           // [256]
    float* __restrict__ out)                  // [B,M,256]
{
    __shared__ float eT[RR * LSTR];   // edge_feat tile [128 x 260], reused as h tile
    __shared__ float xbuf[SSN * DD];
    __shared__ float maskf[RR];
    __shared__ float denom[SSN];
    __shared__ float redS[NWAVE], redQ[NWAVE], redMR[2 * SSN];

    const int bm0  = blockIdx.x * SSN;    // first supernode of this block
    const int b    = bm0 / MM;
    const int m0   = bm0 % MM;
    const int tid  = threadIdx.x;
    const int wave = tid >> 5;
    const int lane = tid & 31;
    const int hi   = lane >> 4;           // half-wave select (K pair / M row group)
    const int lo   = lane & 15;

    const int mh    = wave & 1;           // M half: rows [64*mh, 64*mh+64)
    const int ng    = wave >> 1;          // N group 0..7
    const int n0    = ng * 32;            // columns [n0, n0+32)
    const int mbase = mh * 4;             // first of this wave's 4 M-tiles

    const size_t nbase = (size_t)bm0 * KN;   // neighbor_idx/mask base (rows contiguous)

    // ---------------- Phase 1: gather 128 neighbor rows into LDS ----------------
    {
        const int r  = tid >> 2;            // edge row 0..127
        const int c0 = (tid & 3) * 64;      // quarter row (64 floats) per thread
        const int idx = neighbor_idx[nbase + r];
        const float4* src = (const float4*)(point_feat + ((size_t)b * NN + idx) * DD + c0);
        float* dst = eT + r * LSTR + c0;
        #pragma unroll 4
        for (int i = 0; i < 16; ++i) {
            float4 v = src[i];
            dst[i * 4 + 0] = v.x; dst[i * 4 + 1] = v.y;
            dst[i * 4 + 2] = v.z; dst[i * 4 + 3] = v.w;
        }
    }
    if (tid < RR) {
        const int r   = tid;
        const int s   = r >> 5;
        const int idx = neighbor_idx[nbase + r];
        const float nx = point_xyz[((size_t)b * NN + idx) * 3 + 0];
        const float ny = point_xyz[((size_t)b * NN + idx) * 3 + 1];
        const float nz = point_xyz[((size_t)b * NN + idx) * 3 + 2];
        const float sx = supernode_xyz[((size_t)b * MM + m0 + s) * 3 + 0];
        const float sy = supernode_xyz[((size_t)b * MM + m0 + s) * 3 + 1];
        const float sz = supernode_xyz[((size_t)b * MM + m0 + s) * 3 + 2];
        const float rx = nx - sx, ry = ny - sy, rz = nz - sz;
        float* row = eT + r * LSTR;
        row[256] = rx; row[257] = ry; row[258] = rz;
        row[259] = sqrtf(rx * rx + ry * ry + rz * rz);
        maskf[r] = neighbor_mask[nbase + r] ? 1.0f : 0.0f;
    }
    __syncthreads();

    if (tid < SSN) {
        float s = 0.0f;
        for (int k = 0; k < KN; ++k) s += maskf[tid * KN + k];
        denom[tid] = fmaxf(s, 1.0f);
    }

    // ---------------- GEMM1: h = SiLU(edge_feat @ W1 + b1) ----------------
    const v8f vz = {0.f,0.f,0.f,0.f,0.f,0.f,0.f,0.f};
    v8f acc[2][4];                        // [nt][t], 4 M-tiles of 16 rows
    #pragma unroll
    for (int nt = 0; nt < 2; ++nt)
        #pragma unroll
        for (int t = 0; t < 4; ++t) acc[nt][t] = vz;

    for (int k0 = 0; k0 < DIN; k0 += 4) {
        const int ka = k0 + 2 * hi;       // this half-wave's K pair
        v2f w0, w1;
        const float* wp = W1 + (size_t)ka * DD;
        w0.x = wp[n0 + lo];        w0.y = wp[DD + n0 + lo];
        w1.x = wp[n0 + 16 + lo];   w1.y = wp[DD + n0 + 16 + lo];
        #pragma unroll
        for (int t = 0; t < 4; ++t) {
            const int row = (16 * (mbase + t) + lo) * LSTR;
            v2f a;
            a.x = eT[row + ka];
            a.y = eT[row + ka + 1];
            acc[0][t] = __builtin_amdgcn_wmma_f32_16x16x4_f32(false, a, false, w0, (short)0, acc[0][t], false, false);
            acc[1][t] = __builtin_amdgcn_wmma_f32_16x16x4_f32(false, a, false, w1, (short)0, acc[1][t], false, false);
        }
    }
    __syncthreads();   // all waves done reading edge tile

    // bias + SiLU, overwrite LDS tile with h
    {
        const float bias0 = b1[n0 + lo];
        const float bias1 = b1[n0 + 16 + lo];
        #pragma unroll
        for (int t = 0; t < 4; ++t) {
            #pragma unroll
            for (int v = 0; v < 8; ++v) {
                const int r = 16 * (mbase + t) + v + 8 * hi;
                float h0 = acc[0][t][v] + bias0;
                eT[r * LSTR + n0 + lo]      = h0 / (1.0f + __expf(-h0));
                float h1 = acc[1][t][v] + bias1;
                eT[r * LSTR + n0 + 16 + lo] = h1 / (1.0f + __expf(-h1));
            }
        }
    }
    __syncthreads();

    // ---------------- GEMM2: msg = h @ W2 + b2 ----------------
    #pragma unroll
    for (int nt = 0; nt < 2; ++nt)
        #pragma unroll
        for (int t = 0; t < 4; ++t) acc[nt][t] = vz;

    for (int k0 = 0; k0 < DD; k0 += 4) {
        const int ka = k0 + 2 * hi;
        v2f w0, w1;
        const float* wp = W2 + (size_t)ka * DD;
        w0.x = wp[n0 + lo];        w0.y = wp[DD + n0 + lo];
        w1.x = wp[n0 + 16 + lo];   w1.y = wp[DD + n0 + 16 + lo];
        #pragma unroll
        for (int t = 0; t < 4; ++t) {
            const int row = (16 * (mbase + t) + lo) * LSTR;
            v2f a;
            a.x = eT[row + ka];
            a.y = eT[row + ka + 1];
            acc[0][t] = __builtin_amdgcn_wmma_f32_16x16x4_f32(false, a, false, w0, (short)0, acc[0][t], false, false);
            acc[1][t] = __builtin_amdgcn_wmma_f32_16x16x4_f32(false, a, false, w1, (short)0, acc[1][t], false, false);
        }
    }

    // ---------------- masked mean over K per supernode, residual ----------------
    {
        const float bias0 = b2[n0 + lo];
        const float bias1 = b2[n0 + 16 + lo];
        float ps[2][2];                  // [local supernode][nt]
        ps[0][0] = ps[0][1] = ps[1][0] = ps[1][1] = 0.0f;
        #pragma unroll
        for (int t = 0; t < 4; ++t) {
            const int sl = t >> 1;       // local supernode within this M half
            #pragma unroll
            for (int v = 0; v < 8; ++v) {
                const float mk = maskf[16 * (mbase + t) + v + 8 * hi];
                ps[sl][0] += (acc[0][t][v] + bias0) * mk;
                ps[sl][1] += (acc[1][t][v] + bias1) * mk;
            }
        }
        #pragma unroll
        for (int sl = 0; sl < 2; ++sl) {
            ps[sl][0] += __shfl_xor(ps[sl][0], 16, 32);   // fold the two half-waves
            ps[sl][1] += __shfl_xor(ps[sl][1], 16, 32);
        }
        if (hi == 0) {
            #pragma unroll
            for (int sl = 0; sl < 2; ++sl) {
                const int s = 2 * mh + sl;
                const float inv = 1.0f / denom[s];
                const size_t base = ((size_t)bm0 + s) * DD;
                xbuf[s * DD + n0 + lo]      = init_feat[base + n0 + lo]      + ps[sl][0] * inv;
                xbuf[s * DD + n0 + 16 + lo] = init_feat[base + n0 + 16 + lo] + ps[sl][1] * inv;
            }
        }
    }
    __syncthreads();

    // ---------------- LayerNorm: 4 supernodes x 256 cols, 2 cols per thread ----------------
    {
        const int s  = wave >> 2;        // 4 waves per supernode
        const int lt = tid & 127;        // thread within supernode group
        const float x0 = xbuf[s * DD + lt];
        const float x1 = xbuf[s * DD + 128 + lt];
        float sm = x0 + x1, q = x0 * x0 + x1 * x1;
        #pragma unroll
        for (int off = 16; off > 0; off >>= 1) {
            sm += __shfl_down(sm, off, 32);
            q  += __shfl_down(q,  off, 32);
        }
        if (lane == 0) { redS[wave] = sm; redQ[wave] = q; }
        __syncthreads();
        if (tid < SSN) {
            float S = 0.0f, Q = 0.0f;
            for (int w = 0; w < 4; ++w) { S += redS[tid * 4 + w]; Q += redQ[tid * 4 + w]; }
            const float mu  = S * (1.0f / DD);
            const float var = Q * (1.0f / DD) - mu * mu;
            redMR[tid * 2]     = mu;
            redMR[tid * 2 + 1] = rsqrtf(var + LN_EPS);
        }
        __syncthreads();
        const float mu   = redMR[s * 2];
        const float rstd = redMR[s * 2 + 1];
        const size_t obase = ((size_t)bm0 + s) * DD;
        out[obase + lt]       = (x0 - mu) * rstd * ln_g[lt]       + ln_b[lt];
        out[obase + 128 + lt] = (x1 - mu) * rstd * ln_g[128 + lt] + ln_b[128 + lt];
    }
}

extern "C" void kernel_launch(void* const* d_in, const int* in_sizes, int n_in,
                              void* d_out, int out_size, void* d_ws, size_t ws_size,
                              hipStream_t stream) {
    const float* point_feat    = (const float*)d_in[0];
    const float* point_xyz     = (const float*)d_in[1];
    const float* supernode_xyz = (const float*)d_in[2];
    const int*   neighbor_idx  = (const int*)d_in[3];
    const unsigned char* mask  = (const unsigned char*)d_in[4];
    const float* init_feat     = (const float*)d_in[5];
    const float* W1            = (const float*)d_in[6];
    const float* b1            = (const float*)d_in[7];
    const float* W2            = (const float*)d_in[8];
    const float* b2            = (const float*)d_in[9];
    const float* ln_g          = (const float*)d_in[10];
    const float* ln_b          = (const float*)d_in[11];
    float* out = (float*)d_out;

    dim3 grid((BB * MM) / SSN);   // 4 supernodes per workgroup
    dim3 block(512);              // 16 wave32
    p2s_msg_kernel<<<grid, block, 0, stream>>>(
        point_feat, point_xyz, supernode_xyz, neighbor_idx, mask, init_feat,
        W1, b1, W2, b2, ln_g, ln_b, out);
}